// Self_Attn_19653770346951
// MI455X (gfx1250) — compile-verified
//
#include <hip/hip_runtime.h>

typedef float v2f __attribute__((ext_vector_type(2)));
typedef float v8f __attribute__((ext_vector_type(8)));
typedef unsigned int v4u __attribute__((ext_vector_type(4)));
typedef int v4i __attribute__((ext_vector_type(4)));
typedef int v8i __attribute__((ext_vector_type(8)));

#define WMMA_F32(a, b, c) \
  __builtin_amdgcn_wmma_f32_16x16x4_f32(false, (a), false, (b), (short)0, (c), false, false)

#if defined(__has_builtin)
#if __has_builtin(__builtin_amdgcn_tensor_load_to_lds) && \
    __has_builtin(__builtin_amdgcn_s_wait_tensorcnt)
#define USE_TDM 1
#endif
#endif
#ifndef USE_TDM
#define USE_TDM 0
#endif

constexpr int kB  = 4;
constexpr int kC  = 64;
constexpr int kCk = 8;
constexpr int kN  = 4096;     // 64*64
constexpr int kMC = 64;       // H rows staged per chunk
constexpr int kNCH = kN / kMC;   // 64 chunks
constexpr int kChunkElems = kMC * kC;   // 4096 floats = 16 KB

// ---------------------------------------------------------------------------
// Kernel 1: 1x1-conv projections.  WMMA-friendly layouts:
//   qT : (B, N, 8)   contiguous A-tile float2 loads
//   kM : (B, 8, N)   coalesced B-tile loads
//   hT : (B, N, 64)  coalesced H^T rows (also TDM-friendly: contiguous)
// ---------------------------------------------------------------------------
__global__ __launch_bounds__(256) void proj_kernel(
    const float* __restrict__ x,
    const float* __restrict__ Wq, const float* __restrict__ bq,
    const float* __restrict__ Wk, const float* __restrict__ bk,
    const float* __restrict__ Wh, const float* __restrict__ bh,
    float* __restrict__ qT, float* __restrict__ kM, float* __restrict__ hT)
{
  int g = blockIdx.x * blockDim.x + threadIdx.x;
  if (g >= kB * kN) return;
  int b = g >> 12;
  int n = g & (kN - 1);

  const float* xb = x + (size_t)b * kC * kN + n;
  float xv[kC];
#pragma unroll
  for (int c = 0; c < kC; ++c) xv[c] = xb[(size_t)c * kN];

#pragma unroll 1
  for (int i = 0; i < kCk; ++i) {
    float aq = bq[i], ak = bk[i];
#pragma unroll
    for (int c = 0; c < kC; ++c) {
      aq = fmaf(Wq[i * kC + c], xv[c], aq);
      ak = fmaf(Wk[i * kC + c], xv[c], ak);
    }
    qT[(size_t)g * kCk + i] = aq;
    kM[(size_t)b * kCk * kN + (size_t)i * kN + n] = ak;
  }
#pragma unroll 1
  for (int o = 0; o < kC; ++o) {
    float ah = bh[o];
#pragma unroll
    for (int c = 0; c < kC; ++c) ah = fmaf(Wh[o * kC + c], xv[c], ah);
    hT[(size_t)g * kC + o] = ah;
  }
}

// ---------------------------------------------------------------------------
// TDM: DMA kChunkElems contiguous f32 from global into LDS (1-D tile).
// D# per CDNA5 ISA ch.8: group0 = {count|flags, lds_addr, global_addr, type},
// group1 = {mask|data_size, dims/strides}, groups 2/3 unused (zero).
// ---------------------------------------------------------------------------
#if USE_TDM
__device__ __forceinline__ void tdm_load_chunk(const float* src, unsigned lds_off)
{
  unsigned long long ga = (unsigned long long)(uintptr_t)src;
  v4u g0;
  g0.x = 1u;                                  // count=1, user descriptor
  g0.y = lds_off;                             // lds_addr (bytes)
  g0.z = (unsigned)(ga & 0xffffffffu);        // global_addr[31:0]
  g0.w = (unsigned)(ga >> 32) | 0x80000000u;  // global_addr[56:32] | type=2<<30
  v8i g1;
  g1[0] = 0x00020000;   // workgroup_mask=0, data_size=2 (4B)
  g1[1] = (int)(kChunkElems << 16);   // tensor_dim0[15:0] in [31:16]
  g1[2] = 0x00010000;   // tensor_dim0 hi=0, tensor_dim1=1
  g1[3] = (int)(kChunkElems << 16);   // tensor_dim1 hi=0, tile_dim0
  g1[4] = 1;            // tile_dim1=1, tile_dim2=0
  g1[5] = kChunkElems;  // tensor_dim0_stride lo
  g1[6] = 0;
  g1[7] = 0;
  v4i z4 = {0, 0, 0, 0};
#if __clang_major__ >= 23
  v8i z8 = {0, 0, 0, 0, 0, 0, 0, 0};
  __builtin_amdgcn_tensor_load_to_lds(g0, g1, z4, z4, z8, 0);
#else
  __builtin_amdgcn_tensor_load_to_lds(g0, g1, z4, z4, 0);
#endif
}
#endif

// ---------------------------------------------------------------------------
// S-tile: s[n,m] = sum_c qT[n,c] * k[c,m]  (two 16x16x4 f32 WMMAs)
// ---------------------------------------------------------------------------
__device__ __forceinline__ v8f qk_tile(v2f aq0, v2f aq1,
                                       const float* __restrict__ kb,
                                       int m0, int cl, int hh)
{
  v2f b0, b1;
  b0.x = kb[(size_t)(2 * hh + 0) * kN + m0 + cl];
  b0.y = kb[(size_t)(2 * hh + 1) * kN + m0 + cl];
  b1.x = kb[(size_t)(4 + 2 * hh + 0) * kN + m0 + cl];
  b1.y = kb[(size_t)(4 + 2 * hh + 1) * kN + m0 + cl];
  v8f s = {0.f, 0.f, 0.f, 0.f, 0.f, 0.f, 0.f, 0.f};
  s = WMMA_F32(aq0, b0, s);
  s = WMMA_F32(aq1, b1, s);
  return s;
}

// ---------------------------------------------------------------------------
// Kernel 2: fused attention.  One wave per 16-row n-tile, 4 waves/block
// (same batch, consecutive n-tiles).  H^T is staged per-block into LDS by
// the Tensor Data Mover, double buffered, overlapping pass 1 and compute.
// ---------------------------------------------------------------------------
__global__ __launch_bounds__(128) void attn_kernel(
    const float* __restrict__ qT, const float* __restrict__ kM,
    const float* __restrict__ hT,
    const float* __restrict__ Wv, const float* __restrict__ bv,
    const float* __restrict__ gamma,
    const float* __restrict__ x, float* __restrict__ out)
{
  constexpr int PSTR = 18;   // P tile row stride (even -> aligned b64)
  constexpr int ASTR = 66;   // ah tile row stride (even)
  constexpr int VSTR = 17;   // v tile (c-major) row stride
  __shared__ float lds_h[2][kChunkElems];          // 32 KB: staged H^T chunks
  __shared__ float lds_p[4][16 * PSTR];            // per-wave P tile
  __shared__ float lds_av[4][kC * VSTR];           // per-wave ah/v scratch (reused)

  const int w    = threadIdx.x >> 5;
  const int lane = threadIdx.x & 31;
  const int cl   = lane & 15;
  const int hh   = lane >> 4;

  const int tile = blockIdx.x * 4 + w;      // 0..1023
  const int b    = tile >> 8;               // 256 n-tiles per batch (block-uniform)
  const int n0   = (tile & 255) * 16;

  const float* qb = qT + (size_t)b * kN * kCk;
  const float* kb = kM + (size_t)b * kCk * kN;
  const float* hb = hT + (size_t)b * kN * kC;

#if USE_TDM
  // kick off chunk 0 DMA; it overlaps the entire row-max pass
  if (w == 0) tdm_load_chunk(hb, (unsigned)(uintptr_t)&lds_h[0][0]);
#endif

  // loop-invariant Q A-tiles (rows n0..n0+15, K = 8 channels in two chunks)
  v2f aq0 = *(const v2f*)(qb + (size_t)(n0 + cl) * kCk + 2 * hh);
  v2f aq1 = *(const v2f*)(qb + (size_t)(n0 + cl) * kCk + 4 + 2 * hh);

  // ---------------- pass 1: row max ----------------
  float mx[8];
#pragma unroll
  for (int r = 0; r < 8; ++r) mx[r] = -3.0e38f;
#pragma unroll 1
  for (int mt = 0; mt < kN / 16; ++mt) {
    v8f s = qk_tile(aq0, aq1, kb, mt * 16, cl, hh);
#pragma unroll
    for (int r = 0; r < 8; ++r) mx[r] = fmaxf(mx[r], s[r]);
  }
#pragma unroll
  for (int r = 0; r < 8; ++r) {
    mx[r] = fmaxf(mx[r], __shfl_xor(mx[r], 1, 32));
    mx[r] = fmaxf(mx[r], __shfl_xor(mx[r], 2, 32));
    mx[r] = fmaxf(mx[r], __shfl_xor(mx[r], 4, 32));
    mx[r] = fmaxf(mx[r], __shfl_xor(mx[r], 8, 32));
  }

  // ---------------- pass 2: P = exp(S-max); O = P @ H^T ----------------
  float rs[8];
  v8f o[4];
  const v8f vzero = {0.f, 0.f, 0.f, 0.f, 0.f, 0.f, 0.f, 0.f};
#pragma unroll
  for (int r = 0; r < 8; ++r) rs[r] = 0.f;
#pragma unroll
  for (int ct = 0; ct < 4; ++ct) o[ct] = vzero;

  float* pw = lds_p[w];
#pragma unroll 1
  for (int ch = 0; ch < kNCH; ++ch) {
#if USE_TDM
    if (w == 0) {
      if (ch + 1 < kNCH) {
        tdm_load_chunk(hb + (size_t)(ch + 1) * kChunkElems,
                       (unsigned)(uintptr_t)&lds_h[(ch + 1) & 1][0]);
        __builtin_amdgcn_s_wait_tensorcnt(1);   // chunk ch landed
      } else {
        __builtin_amdgcn_s_wait_tensorcnt(0);
      }
    }
#else
    {
      const float4* src = (const float4*)(hb + (size_t)ch * kChunkElems);
      float4* dst = (float4*)&lds_h[ch & 1][0];
      for (int i = threadIdx.x; i < kChunkElems / 4; i += 128) dst[i] = src[i];
    }
#endif
    __syncthreads();
    const float* hl = &lds_h[ch & 1][0];

#pragma unroll 1
    for (int t = 0; t < kMC / 16; ++t) {
      const int m0 = ch * kMC + t * 16;
      v8f s = qk_tile(aq0, aq1, kb, m0, cl, hh);
#pragma unroll
      for (int r = 0; r < 8; ++r) {
        float p = __expf(s[r] - mx[r]);
        rs[r] += p;
        pw[(r + 8 * hh) * PSTR + cl] = p;     // D layout -> row-major LDS
      }
      asm volatile("s_wait_dscnt 0" ::: "memory");  // same-wave LDS in-order
      v2f ap[4];
#pragma unroll
      for (int kk = 0; kk < 4; ++kk)
        ap[kk] = *(const v2f*)(pw + cl * PSTR + kk * 4 + 2 * hh);  // A layout
#pragma unroll
      for (int kk = 0; kk < 4; ++kk) {
        const float* hrow = hl + (t * 16 + kk * 4 + 2 * hh) * kC;
#pragma unroll
        for (int ct = 0; ct < 4; ++ct) {
          v2f bh2;
          bh2.x = hrow[ct * 16 + cl];
          bh2.y = hrow[kC + ct * 16 + cl];
          o[ct] = WMMA_F32(ap[kk], bh2, o[ct]);
        }
      }
    }
    __syncthreads();   // all waves done with buf[ch&1] before it is re-filled
  }

  // rowsum reduce + normalize
#pragma unroll
  for (int r = 0; r < 8; ++r) {
    rs[r] += __shfl_xor(rs[r], 1, 32);
    rs[r] += __shfl_xor(rs[r], 2, 32);
    rs[r] += __shfl_xor(rs[r], 4, 32);
    rs[r] += __shfl_xor(rs[r], 8, 32);
    rs[r] = 1.0f / rs[r];
  }
#pragma unroll
  for (int ct = 0; ct < 4; ++ct)
#pragma unroll
    for (int r = 0; r < 8; ++r) o[ct][r] *= rs[r];

  // ---------------- fused Wv conv: v^T = ah^T @ Wv^T + bv ----------------
  float* aw = lds_av[w];
#pragma unroll
  for (int ct = 0; ct < 4; ++ct)
#pragma unroll
    for (int r = 0; r < 8; ++r)
      aw[(r + 8 * hh) * ASTR + ct * 16 + cl] = o[ct][r];
  asm volatile("s_wait_dscnt 0" ::: "memory");

  v8f vacc[4];
#pragma unroll
  for (int ct = 0; ct < 4; ++ct) {
    float bvv = bv[ct * 16 + cl];
    v8f t = {bvv, bvv, bvv, bvv, bvv, bvv, bvv, bvv};
    vacc[ct] = t;
  }
#pragma unroll 1
  for (int k0 = 0; k0 < kC; k0 += 4) {
    v2f ap2 = *(const v2f*)(aw + cl * ASTR + k0 + 2 * hh);
#pragma unroll
    for (int ct = 0; ct < 4; ++ct) {
      v2f bw = *(const v2f*)(Wv + (size_t)(ct * 16 + cl) * kC + k0 + 2 * hh);
      vacc[ct] = WMMA_F32(ap2, bw, vacc[ct]);
    }
  }

  // ---------------- transpose v tile via LDS (reusing scratch), store ------
  float* vw = lds_av[w];
#pragma unroll
  for (int ct = 0; ct < 4; ++ct)
#pragma unroll
    for (int r = 0; r < 8; ++r)
      vw[(ct * 16 + cl) * VSTR + (r + 8 * hh)] = vacc[ct][r];
  asm volatile("s_wait_dscnt 0" ::: "memory");

  const float gma   = gamma[0];
  const size_t xoff = (size_t)b * kC * kN;
#pragma unroll 4
  for (int it = 0; it < 32; ++it) {
    int c = 2 * it + hh;
    float vv = vw[c * VSTR + cl];
    size_t idx = xoff + (size_t)c * kN + n0 + cl;
    out[idx] = fmaf(gma, vv, x[idx]);
  }
}

// ---------------------------------------------------------------------------
extern "C" void kernel_launch(void* const* d_in, const int* in_sizes, int n_in,
                              void* d_out, int out_size, void* d_ws, size_t ws_size,
                              hipStream_t stream)
{
  (void)in_sizes; (void)n_in; (void)out_size; (void)ws_size;
  const float* x  = (const float*)d_in[0];
  const float* Wq = (const float*)d_in[1];
  const float* bq = (const float*)d_in[2];
  const float* Wk = (const float*)d_in[3];
  const float* bk = (const float*)d_in[4];
  const float* Wh = (const float*)d_in[5];
  const float* bh = (const float*)d_in[6];
  const float* Wv = (const float*)d_in[7];
  const float* bv = (const float*)d_in[8];
  const float* gm = (const float*)d_in[9];

  float* ws = (float*)d_ws;
  float* qT = ws;                                   // B*N*8
  float* kM = qT + (size_t)kB * kN * kCk;           // B*8*N
  float* hT = kM + (size_t)kB * kCk * kN;           // B*N*64   (total ~5.25 MB)

  proj_kernel<<<(kB * kN) / 256, 256, 0, stream>>>(x, Wq, bq, Wk, bk, Wh, bh,
                                                   qT, kM, hT);
  attn_kernel<<<(kB * (kN / 16)) / 4, 128, 0, stream>>>(qT, kM, hT, Wv, bv, gm,
                                                        x, (float*)d_out);
}